// AnalyticRenderer_35519379538661
// MI455X (gfx1250) — compile-verified
//
#include <hip/hip_runtime.h>
#include <math.h>

typedef float v2f __attribute__((ext_vector_type(2)));
typedef float v8f __attribute__((ext_vector_type(8)));

#define NV  4
#define NE  8
#define DEC 976
#define TPR (DEC / 16)   // 61 16-pixel tiles per image row

// Constant buffer layout in d_ws (floats):
//   [OFF_A , +512)  Amat[v][32][4]  : 8 ellipsoids x 4 padded rows of (Sinv_n @ RTKinv_v), col3 = 0
//   [OFF_CN, +128)  CnC [v][8][4]   : (Cn.x, Cn.y, Cn.z, c)
//   [OFF_G , +64 )  G   [8][8]      : (gxx,gyy,gzz,gxy,gxz,gyz,0,0) with G = S^T S
#define OFF_A  0
#define OFF_CN 512
#define OFF_G  640

__device__ __forceinline__ void inv3(const float m[3][3], float inv[3][3]) {
  float c00 =  m[1][1]*m[2][2] - m[1][2]*m[2][1];
  float c01 =  m[1][2]*m[2][0] - m[1][0]*m[2][2];
  float c02 =  m[1][0]*m[2][1] - m[1][1]*m[2][0];
  float det =  m[0][0]*c00 + m[0][1]*c01 + m[0][2]*c02;
  float id  =  1.0f / det;
  inv[0][0] = c00 * id;
  inv[0][1] = (m[0][2]*m[2][1] - m[0][1]*m[2][2]) * id;
  inv[0][2] = (m[0][1]*m[1][2] - m[0][2]*m[1][1]) * id;
  inv[1][0] = c01 * id;
  inv[1][1] = (m[0][0]*m[2][2] - m[0][2]*m[2][0]) * id;
  inv[1][2] = (m[0][2]*m[1][0] - m[0][0]*m[1][2]) * id;
  inv[2][0] = c02 * id;
  inv[2][1] = (m[0][1]*m[2][0] - m[0][0]*m[2][1]) * id;
  inv[2][2] = (m[0][0]*m[1][1] - m[0][1]*m[1][0]) * id;
}

// 32 threads: one per (view, ellipsoid) pair. Builds all fused constants.
__global__ void renderer_precompute(const float* __restrict__ P,
                                    const float* __restrict__ Mu,
                                    const float* __restrict__ S,
                                    float* __restrict__ cons) {
  int t = threadIdx.x;
  if (t >= NV * NE) return;
  int v = t >> 3, n = t & 7;

  float R[3][3], p3[3];
#pragma unroll
  for (int i = 0; i < 3; ++i) {
#pragma unroll
    for (int j = 0; j < 3; ++j) R[i][j] = P[v*12 + i*4 + j];
    p3[i] = P[v*12 + i*4 + 3];
  }
  float Ri[3][3]; inv3(R, Ri);                 // RTKinv
  float C[3];
#pragma unroll
  for (int i = 0; i < 3; ++i)                  // camera center
    C[i] = -(Ri[i][0]*p3[0] + Ri[i][1]*p3[1] + Ri[i][2]*p3[2]);

  float Sm[3][3];
#pragma unroll
  for (int i = 0; i < 3; ++i)
#pragma unroll
    for (int j = 0; j < 3; ++j) Sm[i][j] = S[n*9 + i*3 + j];
  float Si[3][3]; inv3(Sm, Si);                // Sinv

  // Fused map q = (Sinv @ RTKinv) @ [u,w,1]
  float Mc[3][3];
#pragma unroll
  for (int i = 0; i < 3; ++i)
#pragma unroll
    for (int j = 0; j < 3; ++j)
      Mc[i][j] = Si[i][0]*Ri[0][j] + Si[i][1]*Ri[1][j] + Si[i][2]*Ri[2][j];

  // A-matrix rows (padded to 4 rows per ellipsoid, K padded to 4)
#pragma unroll
  for (int i = 0; i < 4; ++i)
#pragma unroll
    for (int k = 0; k < 4; ++k)
      cons[OFF_A + v*128 + (n*4 + i)*4 + k] = (i < 3 && k < 3) ? Mc[i][k] : 0.0f;

  // Cn = Sinv @ (C - mu), c = |Cn|^2 - 1
  float dx = C[0] - Mu[n*3 + 0];
  float dy = C[1] - Mu[n*3 + 1];
  float dz = C[2] - Mu[n*3 + 2];
  float Cn[3];
#pragma unroll
  for (int i = 0; i < 3; ++i) Cn[i] = Si[i][0]*dx + Si[i][1]*dy + Si[i][2]*dz;
  float cc = Cn[0]*Cn[0] + Cn[1]*Cn[1] + Cn[2]*Cn[2] - 1.0f;
  cons[OFF_CN + v*32 + n*4 + 0] = Cn[0];
  cons[OFF_CN + v*32 + n*4 + 1] = Cn[1];
  cons[OFF_CN + v*32 + n*4 + 2] = Cn[2];
  cons[OFF_CN + v*32 + n*4 + 3] = cc;

  if (v == 0) {   // G = S^T S (view independent)
    float gxx = Sm[0][0]*Sm[0][0] + Sm[1][0]*Sm[1][0] + Sm[2][0]*Sm[2][0];
    float gyy = Sm[0][1]*Sm[0][1] + Sm[1][1]*Sm[1][1] + Sm[2][1]*Sm[2][1];
    float gzz = Sm[0][2]*Sm[0][2] + Sm[1][2]*Sm[1][2] + Sm[2][2]*Sm[2][2];
    float gxy = Sm[0][0]*Sm[0][1] + Sm[1][0]*Sm[1][1] + Sm[2][0]*Sm[2][1];
    float gxz = Sm[0][0]*Sm[0][2] + Sm[1][0]*Sm[1][2] + Sm[2][0]*Sm[2][2];
    float gyz = Sm[0][1]*Sm[0][2] + Sm[1][1]*Sm[1][2] + Sm[2][1]*Sm[2][2];
    cons[OFF_G + n*8 + 0] = gxx; cons[OFF_G + n*8 + 1] = gyy;
    cons[OFF_G + n*8 + 2] = gzz; cons[OFF_G + n*8 + 3] = gxy;
    cons[OFF_G + n*8 + 4] = gxz; cons[OFF_G + n*8 + 5] = gyz;
    cons[OFF_G + n*8 + 6] = 0.0f; cons[OFF_G + n*8 + 7] = 0.0f;
  }
}

// Quadratic per (pixel, ellipsoid): returns segment contribution.
__device__ __forceinline__ float ell_val(float qx, float qy, float qz,
                                         float4 cn, float4 g0, float gxz, float gyz) {
  float d2  = qx*qx + qy*qy + qz*qz;
  float inv = 1.0f / d2;
  float cq  = cn.x*qx + cn.y*qy + cn.z*qz;
  float t   = cq*cq*inv - cn.w;                    // disc/4 in normalized space
  float qGq = g0.x*qx*qx + g0.y*qy*qy + g0.z*qz*qz
            + 2.0f*(g0.w*qx*qy + gxz*qx*qz + gyz*qy*qz);
  float val = 2.0f * sqrtf(fmaxf(t, 0.0f) * qGq * inv);
  return (t > 0.0f) ? val : 0.0f;
}

// One wave per (view, u-row); 61 iterations of 16 pixels.
// Two V_WMMA_F32_16X16X4_F32 per iteration compute q for all 8 ellipsoids x 16 pixels.
__global__ void __launch_bounds__(256)
renderer_main(const float* __restrict__ cons, float* __restrict__ out) {
  const int lane   = threadIdx.x & 31;
  const int waveId = threadIdx.x >> 5;
  const int row    = blockIdx.x * 8 + waveId;      // 0 .. NV*DEC-1 (wave-uniform)
  if (row >= NV * DEC) return;
  const int v = row / DEC;
  const int u = row % DEC;

  const int m  = lane & 15;    // pixel within tile / matrix row within half
  const int hi = lane >> 4;    // lane half selects K pair (A/B) and M rows 8..15 (D)
  const int kb = hi << 1;

  // A matrices: rows 0-15 (ellipsoids 0-3) and rows 16-31 (ellipsoids 4-7)
  const float* Am = cons + OFF_A + v*128;
  v2f a1, a2;
  a1.x = Am[m*4 + kb];          a1.y = Am[m*4 + kb + 1];
  a2.x = Am[(16 + m)*4 + kb];   a2.y = Am[(16 + m)*4 + kb + 1];

  // This lane's 4 ellipsoids (whole-ellipsoid split thanks to 4-row padding)
  const int eA = 2*hi, eB = 2*hi + 1, eC = 4 + 2*hi, eD = 5 + 2*hi;
  const float* cn = cons + OFF_CN + v*32;
  const float* gg = cons + OFF_G;
  float4 cA = *(const float4*)(cn + eA*4);
  float4 cB = *(const float4*)(cn + eB*4);
  float4 cC = *(const float4*)(cn + eC*4);
  float4 cD = *(const float4*)(cn + eD*4);
  float4 gA0 = *(const float4*)(gg + eA*8), gA1 = *(const float4*)(gg + eA*8 + 4);
  float4 gB0 = *(const float4*)(gg + eB*8), gB1 = *(const float4*)(gg + eB*8 + 4);
  float4 gC0 = *(const float4*)(gg + eC*8), gC1 = *(const float4*)(gg + eC*8 + 4);
  float4 gD0 = *(const float4*)(gg + eD*8), gD1 = *(const float4*)(gg + eD*8 + 4);

  const v8f  zero = {0.f,0.f,0.f,0.f,0.f,0.f,0.f,0.f};
  const float b0  = hi ? 1.0f : (float)u;          // B rows: K0=u K1=w (lo) | K2=1 K3=0 (hi)
  const size_t obase = ((size_t)v * DEC + u) * (size_t)DEC;

  for (int it = 0; it < TPR; ++it) {
    v2f b;
    b.x = b0;
    b.y = hi ? 0.0f : (float)(it*16 + m);
    v8f d1 = __builtin_amdgcn_wmma_f32_16x16x4_f32(
        false, a1, false, b, (short)0, zero, false, false);
    v8f d2 = __builtin_amdgcn_wmma_f32_16x16x4_f32(
        false, a2, false, b, (short)0, zero, false, false);

    float acc = 0.0f;
    acc += ell_val(d1[0], d1[1], d1[2], cA, gA0, gA1.x, gA1.y);
    acc += ell_val(d1[4], d1[5], d1[6], cB, gB0, gB1.x, gB1.y);
    acc += ell_val(d2[0], d2[1], d2[2], cC, gC0, gC1.x, gC1.y);
    acc += ell_val(d2[4], d2[5], d2[6], cD, gD0, gD1.x, gD1.y);

    // Combine the two lane-halves (same pixel, disjoint ellipsoid sets).
    float tot = acc + __shfl_xor(acc, 16, 32);
    // Both halves hold the identical total: store from all lanes (no divergence).
    out[obase + (size_t)(it*16 + m)] = tot;
  }
}

extern "C" void kernel_launch(void* const* d_in, const int* in_sizes, int n_in,
                              void* d_out, int out_size, void* d_ws, size_t ws_size,
                              hipStream_t stream) {
  const float* P  = (const float*)d_in[0];   // (4,3,4)
  const float* Mu = (const float*)d_in[1];   // (8,3)
  const float* S  = (const float*)d_in[2];   // (8,3,3)
  float* out  = (float*)d_out;               // (4,976,976)
  float* cons = (float*)d_ws;                // 704 floats used

  renderer_precompute<<<1, 32, 0, stream>>>(P, Mu, S, cons);

  const int rows   = NV * DEC;               // 3904 waves
  const int blocks = rows / 8;               // 256 threads = 8 waves per block
  renderer_main<<<blocks, 256, 0, stream>>>(cons, out);
}